// TriplePairwiseCEFocalLoss_23390391894538
// MI455X (gfx1250) — compile-verified
//
#include <hip/hip_runtime.h>

typedef float  f4  __attribute__((ext_vector_type(4)));
typedef int    i4  __attribute__((ext_vector_type(4)));
typedef float  v2f __attribute__((ext_vector_type(2)));
typedef float  v8f __attribute__((ext_vector_type(8)));

#define BLOCK 256

// ---------------------------------------------------------------------------
// Stage 1: one block per row. Stream scores+mask with non-temporal b128 loads,
// compute per-pair focal loss, masked sum + count, write per-row mean to ws.
// HBM roofline: 256 MB single-touch / 23.3 TB/s ~= 11 us; compute is ~2 trans
// + ~15 VALU per element, well under that with 2M threads in flight.
// ---------------------------------------------------------------------------
__global__ __launch_bounds__(BLOCK)
void focal_row_kernel(const float* __restrict__ scores,
                      const int*   __restrict__ head_pos,
                      const int*   __restrict__ tail_pos,
                      const int*   __restrict__ mask,
                      float*       __restrict__ row_out,
                      int S)
{
    const int r   = blockIdx.x;
    const int tid = threadIdx.x;
    const long long rowoff = (long long)r * (long long)S;
    const float* __restrict__ srow = scores + rowoff;
    const int*   __restrict__ mrow = mask   + rowoff;

    const int   h   = head_pos[r];   // uniform -> scalar load
    const int   t   = tail_pos[r];   // uniform -> scalar load
    const float pos = srow[t];       // uniform -> scalar load

    const float LO = 1e-7f;
    const float HI = (float)(1.0 - 1e-7); // 0.99999988f

    float sum = 0.0f;
    int   cnt = 0;

    for (int idx = tid * 4; idx < S; idx += BLOCK * 4) {
        // hint next chunk of the stream into the hierarchy
        if (idx + BLOCK * 4 < S) {
            __builtin_prefetch(srow + idx + BLOCK * 4, 0, 0);
            __builtin_prefetch(mrow + idx + BLOCK * 4, 0, 0);
        }
        // single-touch 256MB stream (> 192MB L2): non-temporal b128 loads
        f4 sv = __builtin_nontemporal_load((const f4*)(srow + idx));
        i4 mv = __builtin_nontemporal_load((const i4*)(mrow + idx));

#pragma unroll
        for (int e = 0; e < 4; ++e) {
            const int  j    = idx + e;
            const bool negp = (mv[e] == 1) & (j != h) & (j != t);

            const float x  = sv[e] - pos;          // score - pos
            const float ax = fabsf(x);
            const float ee = __expf(-ax);          // exp(-|x|)
            // softplus(x) = max(x,0) + log1p(exp(-|x|))   (logpt = -sp)
            const float sp = fmaxf(x, 0.0f) + log1pf(ee);
            // pt = exp(-sp) = sigmoid(-x), via the same exp:
            const float rr = 1.0f / (1.0f + ee);
            float pt = (x >= 0.0f) ? (ee * rr) : rr;
            pt = fminf(fmaxf(pt, LO), HI);
            const float om = 1.0f - pt;
            const float fl = om * om * sp;         // ALPHA=1, GAMMA=2

            sum += negp ? fl : 0.0f;
            cnt += negp ? 1 : 0;
        }
    }

    // wave32 tree reduction
#pragma unroll
    for (int o = 16; o >= 1; o >>= 1) {
        sum += __shfl_xor(sum, o, 32);
        cnt += __shfl_xor(cnt, o, 32);
    }

    __shared__ float ssum[BLOCK / 32];
    __shared__ int   scnt[BLOCK / 32];
    const int wave = tid >> 5;
    if ((tid & 31) == 0) { ssum[wave] = sum; scnt[wave] = cnt; }
    __syncthreads();

    if (tid == 0) {
        float s = 0.0f; int c = 0;
#pragma unroll
        for (int w = 0; w < BLOCK / 32; ++w) { s += ssum[w]; c += scnt[w]; }
        row_out[r] = (c > 0) ? (s / (float)c) : 0.0f;
    }
}

// ---------------------------------------------------------------------------
// Stage 2: reduce B per-row values -> scalar mean via the matrix pipe.
// V_WMMA_F32_16X16X4_F32 with all-ones B: D[m,n] = sum_k A[m,k] + C[m,n],
// so each issue folds 64 values into the f32 accumulator; column n=0
// (lane 0 rows 0-7, lane 16 rows 8-15) carries the running total.
// Loop control is SCALAR (wave id broadcast via readfirstlane, no per-lane
// guards) so EXEC stays all-ones across every WMMA as the ISA requires, and
// the A fragment is one aligned global_load_b64. Tail (N % 512) is folded
// into thread 0's final scalar sum — fully deterministic.
// ---------------------------------------------------------------------------
__global__ __launch_bounds__(BLOCK)
void reduce_rows_wmma(const float* __restrict__ vals, float* __restrict__ out, int N)
{
    const int tid  = threadIdx.x;
    const int lane = tid & 31;
    const int wave = __builtin_amdgcn_readfirstlane(tid >> 5);  // SGPR wave id

    const int N512 = N & ~511;            // 8 waves x 64 values per sweep

    v8f c = {};
    v2f bones = {1.0f, 1.0f};

    for (int base = wave * 64; base < N512; base += (BLOCK / 32) * 64) {
        v2f a = *(const v2f*)(vals + base + lane * 2);   // global_load_b64
        c = __builtin_amdgcn_wmma_f32_16x16x4_f32(
                false, a, false, bones, (short)0, c, false, false);
    }

    // column n=0 lives in lane 0 (rows 0-7) and lane 16 (rows 8-15)
    float psum = 0.0f;
#pragma unroll
    for (int k = 0; k < 8; ++k) psum += c[k];

    __shared__ float part[(BLOCK / 32) * 2];
    if (lane == 0)  part[wave * 2 + 0] = psum;
    if (lane == 16) part[wave * 2 + 1] = psum;
    __syncthreads();

    if (tid == 0) {
        float tot = 0.0f;
#pragma unroll
        for (int i = 0; i < (BLOCK / 32) * 2; ++i) tot += part[i];
        for (int i = N512; i < N; ++i) tot += vals[i];   // scalar tail
        out[0] = tot / (float)N;
    }
}

// ---------------------------------------------------------------------------
extern "C" void kernel_launch(void* const* d_in, const int* in_sizes, int n_in,
                              void* d_out, int out_size, void* d_ws, size_t ws_size,
                              hipStream_t stream)
{
    const float* scores = (const float*)d_in[0];
    const int*   head   = (const int*)  d_in[1];
    const int*   tail   = (const int*)  d_in[2];
    const int*   mask   = (const int*)  d_in[3];
    float*       out    = (float*)d_out;
    float*       rowbuf = (float*)d_ws;

    const int Bn = in_sizes[1];              // 8192
    const int S  = in_sizes[0] / Bn;         // 4096

    focal_row_kernel<<<Bn, BLOCK, 0, stream>>>(scores, head, tail, mask, rowbuf, S);
    reduce_rows_wmma<<<1, BLOCK, 0, stream>>>(rowbuf, out, Bn);
}